// CartBondedWholePoseScoringModule_12180527251547
// MI455X (gfx1250) — compile-verified
//
#include <hip/hip_runtime.h>
#include <hip/hip_bf16.h>

// ---------------------------------------------------------------------------
// CartBonded whole-pose scoring for MI455X (gfx1250, wave32).
//  - bond/angle/torsion kernels: 1 thread per term, NT streamed loads,
//    L2-resident coord gathers, LDS 64-bin per-pose accumulation,
//    spread global f32 atomics into 256 slots in d_ws.
//  - final reduction (256 slots x 64 poses) done with chained
//    V_WMMA_F32_16X16X4_F32 (B = ones => layout-robust row sums).
// ---------------------------------------------------------------------------

#define NPOSES      64
#define POSE_SHIFT  14        // MAX_ATOMS = 16384 = 2^14
#define NSLOTS      256       // spread slots for global atomics (64 KB in d_ws)
#define EPSF        1e-12f
#define TBLOCK      256
#define MAXBLK      2048

typedef float v2f __attribute__((ext_vector_type(2)));
typedef float v8f __attribute__((ext_vector_type(8)));
typedef int   v2i __attribute__((ext_vector_type(2)));
typedef int   v4i __attribute__((ext_vector_type(4)));

struct F3 { float x, y, z; };
__device__ __forceinline__ F3 f3sub(F3 a, F3 b) { return {a.x - b.x, a.y - b.y, a.z - b.z}; }
__device__ __forceinline__ F3 f3cross(F3 a, F3 b) {
  return {a.y * b.z - a.z * b.y, a.z * b.x - a.x * b.z, a.x * b.y - a.y * b.x};
}
__device__ __forceinline__ float f3dot(F3 a, F3 b) { return a.x * b.x + a.y * b.y + a.z * b.z; }
__device__ __forceinline__ F3 ldc(const float* __restrict__ c, int idx) {
  const float* p = c + 3 * idx;     // coords are L2-resident (12.6 MB << 192 MB)
  return {p[0], p[1], p[2]};
}

// ---------------------------------------------------------------------------
__global__ void zero_ws_kernel(float* __restrict__ ws) {
  int i = blockIdx.x * blockDim.x + threadIdx.x;
  if (i < NSLOTS * NPOSES) ws[i] = 0.0f;
}

// ---------------------------------------------------------------------------
__global__ void bond_kernel(const float* __restrict__ coords,
                            const float* __restrict__ Kp,
                            const float* __restrict__ x0,
                            const int*   __restrict__ atoms,   // [n][2]
                            const int*   __restrict__ pidx,
                            float*       __restrict__ ws, int n) {
  __shared__ float acc[NPOSES];
  if (threadIdx.x < NPOSES) acc[threadIdx.x] = 0.0f;
  __syncthreads();

  const int stride = gridDim.x * blockDim.x;
  for (int i = blockIdx.x * blockDim.x + threadIdx.x; i < n; i += stride) {
    v2i   a  = __builtin_nontemporal_load(((const v2i*)atoms) + i);  // b64, TH=NT
    float xv = __builtin_nontemporal_load(x0 + i);
    int   pp = __builtin_nontemporal_load(pidx + i);
    float K  = Kp[pp];
    F3 p0 = ldc(coords, a.x);
    F3 p1 = ldc(coords, a.y);
    F3 dv = f3sub(p0, p1);
    float d = sqrtf(f3dot(dv, dv) + EPSF);
    float t = d - xv;
    atomicAdd(&acc[a.x >> POSE_SHIFT], K * t * t);                   // ds_add_f32
  }
  __syncthreads();
  if (threadIdx.x < NPOSES)
    atomicAdd(&ws[(blockIdx.x & (NSLOTS - 1)) * NPOSES + threadIdx.x], acc[threadIdx.x]);
}

// ---------------------------------------------------------------------------
__global__ void angle_kernel(const float* __restrict__ coords,
                             const float* __restrict__ Kp,
                             const float* __restrict__ x0,
                             const int*   __restrict__ atoms,  // [n][3]
                             const int*   __restrict__ pidx,
                             float*       __restrict__ ws, int n) {
  __shared__ float acc[NPOSES];
  if (threadIdx.x < NPOSES) acc[threadIdx.x] = 0.0f;
  __syncthreads();

  const int stride = gridDim.x * blockDim.x;
  for (int i = blockIdx.x * blockDim.x + threadIdx.x; i < n; i += stride) {
    const int* arow = atoms + 3 * i;          // 12B stride -> three b32 NT loads
    int a0 = __builtin_nontemporal_load(arow + 0);
    int a1 = __builtin_nontemporal_load(arow + 1);
    int a2 = __builtin_nontemporal_load(arow + 2);
    float xv = __builtin_nontemporal_load(x0 + i);
    int   pp = __builtin_nontemporal_load(pidx + i);
    float K  = Kp[pp];
    F3 p0 = ldc(coords, a0);
    F3 p1 = ldc(coords, a1);
    F3 p2 = ldc(coords, a2);
    F3 u = f3sub(p0, p1);
    F3 v = f3sub(p2, p1);
    F3 cx = f3cross(u, v);
    float theta = atan2f(sqrtf(f3dot(cx, cx) + EPSF), f3dot(u, v));
    float t = theta - xv;
    atomicAdd(&acc[a0 >> POSE_SHIFT], K * t * t);
  }
  __syncthreads();
  if (threadIdx.x < NPOSES)
    atomicAdd(&ws[(blockIdx.x & (NSLOTS - 1)) * NPOSES + threadIdx.x], acc[threadIdx.x]);
}

// ---------------------------------------------------------------------------
__global__ void torsion_kernel(const float* __restrict__ coords,
                               const float* __restrict__ Kp,
                               const float* __restrict__ x0,
                               const int*   __restrict__ atoms,  // [n][4]
                               const int*   __restrict__ pidx,
                               float*       __restrict__ ws, int n) {
  __shared__ float acc[NPOSES];
  if (threadIdx.x < NPOSES) acc[threadIdx.x] = 0.0f;
  __syncthreads();

  const int stride = gridDim.x * blockDim.x;
  for (int i = blockIdx.x * blockDim.x + threadIdx.x; i < n; i += stride) {
    v4i   a  = __builtin_nontemporal_load(((const v4i*)atoms) + i);  // b128, TH=NT
    float xv = __builtin_nontemporal_load(x0 + i);
    int   pp = __builtin_nontemporal_load(pidx + i);
    // next grid-stride tile is ~8 MB away: pull its line early (global_prefetch_b8)
    __builtin_prefetch(((const v4i*)atoms) + i + stride, 0, 0);
    float K = Kp[pp];
    F3 p0 = ldc(coords, a.x);
    F3 p1 = ldc(coords, a.y);
    F3 p2 = ldc(coords, a.z);
    F3 p3 = ldc(coords, a.w);
    F3 b1 = f3sub(p1, p0);
    F3 b2 = f3sub(p2, p1);
    F3 b3 = f3sub(p3, p2);
    F3 n1 = f3cross(b1, b2);
    F3 n2 = f3cross(b2, b3);
    float inv = 1.0f / sqrtf(f3dot(b2, b2) + EPSF);
    F3 b2n = {b2.x * inv, b2.y * inv, b2.z * inv};
    F3 m1 = f3cross(n1, b2n);
    float phi = atan2f(f3dot(m1, n2), f3dot(n1, n2));
    float e = K * (1.0f + cosf(3.0f * phi - xv));
    atomicAdd(&acc[a.x >> POSE_SHIFT], e);
  }
  __syncthreads();
  if (threadIdx.x < NPOSES)
    atomicAdd(&ws[(blockIdx.x & (NSLOTS - 1)) * NPOSES + threadIdx.x], acc[threadIdx.x]);
}

// ---------------------------------------------------------------------------
// out[p] = sum_{s<256} ws[s][p] via chained V_WMMA_F32_16X16X4_F32.
// B = all-ones => D[m][n] = sum_k A[m][k] + C[m][n]; result is invariant to
// B's lane layout and to K-permutation inside each 4-slot chunk, so only the
// documented D row mapping (VGPR r: lanes 0-15 -> M=r, lanes 16-31 -> M=r+8)
// is load-bearing. 1 block, 128 threads = 4 waves, 16 poses per wave.
__global__ void reduce_wmma_kernel(const float* __restrict__ ws,
                                   float* __restrict__ out) {
  const int lane = threadIdx.x & 31;
  const int wave = threadIdx.x >> 5;            // 0..3
  const int pose = wave * 16 + (lane & 15);     // A rows M=0..15 in both halves
  const int klo  = (lane & 16) ? 2 : 0;         // A VGPR0: K=0 | K=2, VGPR1: K=1 | K=3

  v8f c = {};
  v2f b; b.x = 1.0f; b.y = 1.0f;                // ones matrix

  for (int k0 = 0; k0 < NSLOTS; k0 += 4) {
    v2f a;
    a.x = ws[(k0 + klo + 0) * NPOSES + pose];
    a.y = ws[(k0 + klo + 1) * NPOSES + pose];
    c = __builtin_amdgcn_wmma_f32_16x16x4_f32(false, a, false, b,
                                              (short)0, c, false, false);
  }

  // D layout: VGPR r holds M=r (lanes 0-15, N=lane) and M=r+8 (lanes 16-31).
  // Lane 0 -> N=0 rows 0..7; lane 16 -> N=0 rows 8..15.
  if ((lane & 15) == 0) {
    int mbase = wave * 16 + ((lane & 16) ? 8 : 0);
#pragma unroll
    for (int r = 0; r < 8; ++r) out[mbase + r] = c[r];
  }
}

// ---------------------------------------------------------------------------
static inline int grid_for(int n) {
  int b = (n + TBLOCK - 1) / TBLOCK;
  return b < MAXBLK ? (b > 0 ? b : 1) : MAXBLK;
}

extern "C" void kernel_launch(void* const* d_in, const int* in_sizes, int n_in,
                              void* d_out, int out_size, void* d_ws, size_t ws_size,
                              hipStream_t stream) {
  const float* coords = (const float*)d_in[0];
  const float* Kp     = (const float*)d_in[1];
  const float* bx0    = (const float*)d_in[2];
  const float* ax0    = (const float*)d_in[3];
  const float* tx0    = (const float*)d_in[4];
  const int*   batoms = (const int*)d_in[5];
  const int*   bpar   = (const int*)d_in[6];
  const int*   aatoms = (const int*)d_in[7];
  const int*   apar   = (const int*)d_in[8];
  const int*   tatoms = (const int*)d_in[9];
  const int*   tpar   = (const int*)d_in[10];

  const int nb = in_sizes[2];
  const int na = in_sizes[3];
  const int nt = in_sizes[4];

  float* ws  = (float*)d_ws;     // first 64 KB: [NSLOTS][NPOSES] partials
  float* out = (float*)d_out;

  zero_ws_kernel<<<(NSLOTS * NPOSES + TBLOCK - 1) / TBLOCK, TBLOCK, 0, stream>>>(ws);
  bond_kernel   <<<grid_for(nb), TBLOCK, 0, stream>>>(coords, Kp, bx0, batoms, bpar, ws, nb);
  angle_kernel  <<<grid_for(na), TBLOCK, 0, stream>>>(coords, Kp, ax0, aatoms, apar, ws, na);
  torsion_kernel<<<grid_for(nt), TBLOCK, 0, stream>>>(coords, Kp, tx0, tatoms, tpar, ws, nt);
  reduce_wmma_kernel<<<1, 128, 0, stream>>>(ws, out);
}